// RoiPooling_29386166239332
// MI455X (gfx1250) — compile-verified
//
#include <hip/hip_runtime.h>
#include <math.h>
#include <stdint.h>

#define FH 56
#define FW 56
#define FC 256
#define NBINS 21                 // 1 + 4 + 16
#define MAXCOLS 32               // max bin width is <= 31 columns
#define TILE_ELEMS (MAXCOLS * FC)
#define FEAT_ELEMS (FH * FW * FC)

#if defined(__gfx1250__) &&                                                    \
    __has_builtin(__builtin_amdgcn_global_load_async_to_lds_b128) &&           \
    __has_builtin(__builtin_amdgcn_s_wait_asynccnt)
#define USE_ASYNC 1
#else
#define USE_ASYNC 0
#endif

#if USE_ASYNC
typedef int v4i __attribute__((ext_vector_type(4)));
typedef __attribute__((address_space(1))) v4i* g_v4i_ptr;
typedef __attribute__((address_space(3))) v4i* l_v4i_ptr;

__device__ __forceinline__ g_v4i_ptr to_global_v4(const void* p) {
  // AS1 pointer value == 64-bit global VA
  return (g_v4i_ptr)(uintptr_t)p;
}
__device__ __forceinline__ l_v4i_ptr to_lds_v4(const void* p) {
  // low 32 bits of a generic LDS address are the LDS byte offset
  return (l_v4i_ptr)(uintptr_t)(uint32_t)(uintptr_t)p;
}
#endif

__global__ __launch_bounds__(256) void roi_pool_kernel(
    const float* __restrict__ img,   // [56,56,256] NHWC
    const float* __restrict__ rois,  // [R,4] = x,y,w,h
    float* __restrict__ out) {       // [R,21,256]
  const int tid = (int)threadIdx.x;      // channel
  const int blk = (int)blockIdx.x;       // r*21 + bin
  const int r   = blk / NBINS;
  const int bin = blk - r * NBINS;

  // decode (level n, x-bin bi, y-bin bj); within a level flat index = bi*n + bj
  int n, bi, bj;
  if (bin == 0)      { n = 1; bi = 0; bj = 0; }
  else if (bin < 5)  { n = 2; int k = bin - 1; bi = k >> 1; bj = k & 1; }
  else               { n = 4; int k = bin - 5; bi = k >> 2; bj = k & 3; }

  const float x = rois[r * 4 + 0];
  const float y = rois[r * 4 + 1];
  const float w = rois[r * 4 + 2];
  const float h = rois[r * 4 + 3];

  // Faithful to reference: columns stride by h/n, rows stride by w/n.
  // Use explicit _rn ops (no FMA contraction) + rintf (round-half-even) to
  // bit-match jnp float32 arithmetic.
  const float fn = (float)n;
  const float cl = __fdiv_rn(h, fn);
  const float rl = __fdiv_rn(w, fn);
  const float xa = __fadd_rn(x, __fmul_rn((float)bi, cl));
  const float ya = __fadd_rn(y, __fmul_rn((float)bj, rl));
  int x1 = (int)rintf(xa);
  int x2 = (int)rintf(__fadd_rn(xa, cl));
  int y1 = (int)rintf(ya);
  int y2 = (int)rintf(__fadd_rn(ya, rl));
  x1 = x1 < 0 ? 0 : x1;   y1 = y1 < 0 ? 0 : y1;
  x2 = x2 > FW ? FW : x2; y2 = y2 > FH ? FH : y2;
  const int ncols = x2 - x1;
  const int nrows = y2 - y1;

  float m = -__builtin_inff();

#if USE_ASYNC
  __shared__ float tile[2][TILE_ELEMS];   // 2 x 32KB double buffer

  // Stage one padded row segment (MAXCOLS cols x 256 ch, contiguous in
  // global memory) into LDS: 8 uniform b128 async issues per wave.
  auto issue_row = [&](int rowIdx, int buf) {
    const int rr = y1 + rowIdx;
    const size_t base = (size_t)(rr * FW + x1) * FC;  // element offset
    float* lb = &tile[buf][0];
#pragma unroll
    for (int it = 0; it < (MAXCOLS * FC / 4) / 256; ++it) {  // 8
      const int u = it * 256 + tid;                 // 16-byte unit index
      size_t goff = base + (size_t)u * 4;
      if (goff > (size_t)(FEAT_ELEMS - 4)) goff = (size_t)(FEAT_ELEMS - 4);
      __builtin_amdgcn_global_load_async_to_lds_b128(
          to_global_v4(img + goff), to_lds_v4(lb + (size_t)u * 4), 0, 0);
    }
  };

  if (nrows > 0) issue_row(0, 0);
  for (int ri = 0; ri < nrows; ++ri) {
    const int  buf      = ri & 1;
    const bool has_next = (ri + 1) < nrows;
    if (has_next) issue_row(ri + 1, buf ^ 1);       // prefetch next row
    if (has_next) __builtin_amdgcn_s_wait_asynccnt(8);  // current row done
    else          __builtin_amdgcn_s_wait_asynccnt(0);
    __syncthreads();                                // publish tile to all waves
    const float* lb = &tile[buf][0];
    for (int cc = 0; cc < ncols; ++cc)
      m = fmaxf(m, lb[cc * FC + tid]);
    __syncthreads();                                // done reading before reuse
  }
#else
  // Fallback: direct coalesced global loads (L2-resident feature map).
  for (int rr = y1; rr < y2; ++rr) {
    const float* rowp = img + (size_t)(rr * FW) * FC + tid;
    for (int cc = x1; cc < x2; ++cc)
      m = fmaxf(m, rowp[(size_t)cc * FC]);
  }
#endif

  out[((size_t)r * NBINS + bin) * FC + tid] = m;
}

extern "C" void kernel_launch(void* const* d_in, const int* in_sizes, int n_in,
                              void* d_out, int out_size, void* d_ws, size_t ws_size,
                              hipStream_t stream) {
  (void)n_in; (void)out_size; (void)d_ws; (void)ws_size;
  const float* img  = (const float*)d_in[0];
  const float* rois = (const float*)d_in[1];
  float* out = (float*)d_out;
  const int R = in_sizes[1] / 4;   // 32
  dim3 grid((unsigned)(R * NBINS));
  dim3 block(256);
  hipLaunchKernelGGL(roi_pool_kernel, grid, block, 0, stream, img, rois, out);
}